// IIRConv2d_79654463471800
// MI455X (gfx1250) — compile-verified
//
#include <hip/hip_runtime.h>
#include <stdint.h>

// Problem constants (fixed by the reference)
#define BATCH     4
#define CHANNELS  64
#define HEIGHT    512
#define WIDTH     512
#define K1        5
#define NFB       3
#define ROWS_TOTAL (BATCH*CHANNELS*HEIGHT)   // 131072 independent rows

// Kernel tiling
#define WAVES_PER_BLOCK 4
#define THREADS   (WAVES_PER_BLOCK*32)       // 128
#define ROWS_PER_BLOCK THREADS               // one row per thread
#define CHUNK     32                         // columns staged per chunk
#define NCHUNK    (WIDTH/CHUNK)              // 16
#define LDS_STRIDE 36                        // floats per row in LDS (144B: 16B-aligned, padded)
#define BUF_FLOATS (32*LDS_STRIDE)           // per-wave per-buffer: 1152 floats (4608B)

typedef float v4f __attribute__((ext_vector_type(4)));

// Issue one chunk (32 rows x 32 cols) of this wave's rows: 8 x async B128 per wave.
// Global side: lanes cover consecutive 16B granules -> full 128B bursts, marked
// non-temporal (data is streamed exactly once; footprint 256MB > 192MB L2).
// LDS side: per-lane scattered addresses implement the padded row stride.
__device__ __forceinline__ void issue_chunk(const float* __restrict__ xw0,
                                            int ck, uint32_t lds_base_bytes, int lane) {
#pragma unroll
    for (int i = 0; i < 8; ++i) {
        int g  = i * 32 + lane;        // granule id within tile (16B granules)
        int rl = g >> 3;               // local row 0..31 (8 granules per 32-col row)
        int c4 = g & 7;                // granule within row
        const float* gp = xw0 + (size_t)rl * WIDTH + (size_t)ck * CHUNK + (size_t)c4 * 4;
        uint32_t lb = lds_base_bytes + (uint32_t)(rl * LDS_STRIDE + c4 * 4) * 4u;
        asm volatile("global_load_async_to_lds_b128 %0, %1, off th:TH_LOAD_NT"
                     :: "v"(lb), "v"((uint64_t)(uintptr_t)gp)
                     : "memory");
    }
}

__global__ __launch_bounds__(THREADS)
void iir_conv2d_rows(const float* __restrict__ x,
                     const float* __restrict__ w1g,   // (C,1,5)
                     const float* __restrict__ w2g,   // (C,3)
                     float* __restrict__ y)
{
    const int tid  = threadIdx.x;
    const int lane = tid & 31;
    const int wv   = tid >> 5;
    const int R    = blockIdx.x * ROWS_PER_BLOCK + tid;     // this thread's row
    const int Rw0  = blockIdx.x * ROWS_PER_BLOCK + wv * 32; // wave's first row
    const int c    = (R >> 9) & (CHANNELS - 1);             // channel of this row

    // FIR taps (cross-correlation, SAME pad=2) and IIR feedback taps
    const float w0 = w1g[c*K1+0], w1 = w1g[c*K1+1], w2 = w1g[c*K1+2],
                w3 = w1g[c*K1+3], w4 = w1g[c*K1+4];
    const float a0 = w2g[c*NFB+0], a1 = w2g[c*NFB+1], a2 = w2g[c*NFB+2];

    __shared__ float smem[WAVES_PER_BLOCK * 2 * BUF_FLOATS];   // 36,864 B
    const uint32_t wave_lds0 =
        (uint32_t)(uintptr_t)&smem[wv * 2 * BUF_FLOATS];       // flat addr low 32b == LDS offset
    const uint32_t wave_lds1 = wave_lds0 + BUF_FLOATS * 4u;

    const float* xw0  = x + (size_t)Rw0 * WIDTH;   // wave's 32 contiguous rows
    float*       yrow = y + (size_t)R   * WIDTH;

    // Prime the double-buffer pipeline
    issue_chunk(xw0, 0, wave_lds0, lane);
    issue_chunk(xw0, 1, wave_lds1, lane);

    // Causal pipeline with 2-sample delay: at read index j we emit o = j-2.
    // f[o] = w0*x[o-2]+w1*x[o-1]+w2*x[o]+w3*x[o+1]+w4*x[o+2]
    //      = w0*x4 + w1*x3 + w2*x2 + w3*x1 + w4*x[j]
    float x1 = 0.f, x2 = 0.f, x3 = 0.f, x4 = 0.f;   // x[j-1..j-4]
    float y1 = 0.f, y2 = 0.f, y3 = 0.f;             // y[o-1..o-3]
    float yb0 = 0.f, yb1 = 0.f, yb2 = 0.f, yb3 = 0.f;

    const v4f* bv0 = (const v4f*)&smem[wv * 2 * BUF_FLOATS + lane * LDS_STRIDE];
    const v4f* bv1 = (const v4f*)&smem[wv * 2 * BUF_FLOATS + BUF_FLOATS + lane * LDS_STRIDE];

    for (int ck = 0; ck < NCHUNK; ++ck) {
        // Async loads complete in order: <=8 outstanding means chunk ck has landed.
        if (ck < NCHUNK - 1) asm volatile("s_wait_asynccnt 8" ::: "memory");
        else                 asm volatile("s_wait_asynccnt 0" ::: "memory");

        const v4f* bv = (ck & 1) ? bv1 : bv0;
#pragma unroll
        for (int q = 0; q < 8; ++q) {
            v4f xq = bv[q];                     // ds_load_b128, 16B aligned
            float xs[4] = {xq.x, xq.y, xq.z, xq.w};
#pragma unroll
            for (int u = 0; u < 4; ++u) {
                const int t  = q * 4 + u;
                const float xj = xs[u];
                const float fv = w0*x4 + w1*x3 + w2*x2 + w3*x1 + w4*xj;
                const float yv = fv - (a0*y3 + a1*y2 + a2*y1);
                x4 = x3; x3 = x2; x2 = x1; x1 = xj;
                const int o = ck * CHUNK + t - 2;
                if (o >= 0) {
                    y3 = y2; y2 = y1; y1 = yv;
                    switch ((t + 2) & 3) {      // == o & 3 (CHUNK multiple of 4)
                        case 0: yb0 = yv; break;
                        case 1: yb1 = yv; break;
                        case 2: yb2 = yv; break;
                        default: {
                            yb3 = yv;
                            v4f o4 = {yb0, yb1, yb2, yb3};
                            __builtin_nontemporal_store(o4, (v4f*)(yrow + (o - 3)));
                        }
                    }
                }
            }
        }
        // Ensure our ds reads of this buffer retired before the async engine rewrites it.
        if (ck + 2 < NCHUNK) {
            asm volatile("s_wait_dscnt 0" ::: "memory");
            issue_chunk(xw0, ck + 2, (ck & 1) ? wave_lds1 : wave_lds0, lane);
        }
    }

    // Epilogue: j = 512, 513 with zero-padded x -> outputs o = 510, 511
#pragma unroll
    for (int e = 0; e < 2; ++e) {
        const float xj = 0.f;
        const float fv = w0*x4 + w1*x3 + w2*x2 + w3*x1 + w4*xj;
        const float yv = fv - (a0*y3 + a1*y2 + a2*y1);
        x4 = x3; x3 = x2; x2 = x1; x1 = xj;
        y3 = y2; y2 = y1; y1 = yv;
        const int o = WIDTH - 2 + e;            // 510 (o&3==2), 511 (o&3==3)
        if ((o & 3) == 2) {
            yb2 = yv;
        } else {
            yb3 = yv;
            v4f o4 = {yb0, yb1, yb2, yb3};
            __builtin_nontemporal_store(o4, (v4f*)(yrow + (o - 3)));
        }
    }
}

extern "C" void kernel_launch(void* const* d_in, const int* in_sizes, int n_in,
                              void* d_out, int out_size, void* d_ws, size_t ws_size,
                              hipStream_t stream) {
    const float* x  = (const float*)d_in[0];   // (4,64,512,512) f32
    const float* w1 = (const float*)d_in[1];   // (64,1,5) f32
    const float* w2 = (const float*)d_in[2];   // (64,3) f32
    float* yout     = (float*)d_out;           // (4,64,512,512) f32

    dim3 grid(ROWS_TOTAL / ROWS_PER_BLOCK);    // 1024 blocks
    dim3 block(THREADS);                       // 128 threads = 4 waves
    hipLaunchKernelGGL(iir_conv2d_rows, grid, block, 0, stream, x, w1, w2, yout);
}